// HeteroGATLayerLossyReal_46136538693994
// MI455X (gfx1250) — compile-verified
//
#include <hip/hip_runtime.h>
#include <hip/hip_bf16.h>

typedef __attribute__((ext_vector_type(2))) float v2f;
typedef __attribute__((ext_vector_type(8))) float v8f;

#define DIN   256
#define DTOT  256   // H * DOUT
#define HH    4
#define DOUTD 64
#define SLOPE 0.2f

__device__ __forceinline__ float leaky(float x) { return x >= 0.f ? x : SLOPE * x; }

// ---------------------------------------------------------------------------
// Y[M, 256] = X[M, 256] @ W[256, 256]^T + b        (FP32 WMMA 16x16x4)
// One wave per 16x16 output tile; 64 WMMA steps over K=256.
// A layout (32-bit 16x4): lanes 0-15 hold K=k,k+1 ; lanes 16-31 hold K=k+2,k+3
// B[k][n] = W[n, k]  -> same 2-consecutive-float load pattern on W rows.
// ---------------------------------------------------------------------------
__global__ __launch_bounds__(256) void gemm_wmma_f32(
    const float* __restrict__ X, const float* __restrict__ W,
    const float* __restrict__ bias, float* __restrict__ Y, int M) {
  const int lane = threadIdx.x & 31;
  const int wave = threadIdx.x >> 5;
  const int tile = blockIdx.x * 8 + wave;
  const int mtiles = (M + 15) >> 4;
  if (tile >= mtiles * 16) return;           // uniform per wave
  const int tn   = tile & 15;                // 256/16 = 16 N tiles
  const int tm   = tile >> 4;
  const int half = lane >> 4;
  const int l15  = lane & 15;

  int row  = tm * 16 + l15;
  int arow = row < M ? row : (M - 1);        // clamp so loads stay in-bounds
  const float* Ap = X + (size_t)arow * DIN + 2 * half;
  const float* Bp = W + (size_t)(tn * 16 + l15) * DIN + 2 * half;

  v8f acc = {};
#pragma unroll 8
  for (int k = 0; k < DIN; k += 4) {
    v2f a = *(const v2f*)(Ap + k);
    v2f b = *(const v2f*)(Bp + k);
    acc = __builtin_amdgcn_wmma_f32_16x16x4_f32(
        /*neg_a=*/false, a, /*neg_b=*/false, b,
        /*c_mod=*/(short)0, acc, /*reuse_a=*/false, /*reuse_b=*/false);
  }

  float bv = bias[tn * 16 + l15];
#pragma unroll
  for (int v = 0; v < 8; ++v) {
    int r = tm * 16 + v + 8 * half;          // C/D layout: VGPR v -> M = v + 8*half
    if (r < M) Y[(size_t)r * DTOT + tn * 16 + l15] = acc[v] + bv;
  }
}

// ---------------------------------------------------------------------------
// out[n*4 + h] = sum_d Wh[n, h*64 + d] * att[h*64 + d]
// ---------------------------------------------------------------------------
__global__ __launch_bounds__(256) void scores_kernel(
    const float* __restrict__ Wh, const float* __restrict__ att,
    float* __restrict__ out, int N) {
  int t = blockIdx.x * 256 + threadIdx.x;
  if (t >= N * HH) return;
  int n = t >> 2, h = t & 3;
  const float4* a = (const float4*)(Wh + (size_t)n * DTOT + h * DOUTD);
  const float4* b = (const float4*)(att + h * DOUTD);
  float s = 0.f;
#pragma unroll
  for (int i = 0; i < 16; ++i) {
    float4 av = a[i], bv = b[i];
    s += av.x * bv.x + av.y * bv.y + av.z * bv.z + av.w * bv.w;
  }
  out[t] = s;
}

// ---------------------------------------------------------------------------
// Per-edge: softmax over 4 heads of leaky(s_src[src]+s_dst[dst]), then
// acc[dst] += Wh[src] * attn   (wave per edge, lane moves 8 floats of 1 head)
// ---------------------------------------------------------------------------
__global__ __launch_bounds__(256) void edge_agg_kernel(
    const float* __restrict__ Wh, const float* __restrict__ ssrc,
    const float* __restrict__ sdst, const int* __restrict__ esrc,
    const int* __restrict__ edst, float* __restrict__ acc, int E) {
  int e = blockIdx.x * 8 + (threadIdx.x >> 5);
  if (e >= E) return;
  int lane = threadIdx.x & 31;
  int s = esrc[e], d = edst[e];

  float w0 = leaky(ssrc[s * 4 + 0] + sdst[d * 4 + 0]);
  float w1 = leaky(ssrc[s * 4 + 1] + sdst[d * 4 + 1]);
  float w2 = leaky(ssrc[s * 4 + 2] + sdst[d * 4 + 2]);
  float w3 = leaky(ssrc[s * 4 + 3] + sdst[d * 4 + 3]);
  float mx = fmaxf(fmaxf(w0, w1), fmaxf(w2, w3));
  w0 = __expf(w0 - mx); w1 = __expf(w1 - mx);
  w2 = __expf(w2 - mx); w3 = __expf(w3 - mx);
  float inv = 1.f / (w0 + w1 + w2 + w3);
  float a4[4] = {w0 * inv, w1 * inv, w2 * inv, w3 * inv};
  float aw = a4[lane >> 3];                  // lane*8 .. lane*8+7 all in head lane>>3

  const float4* wp = (const float4*)(Wh + (size_t)s * DTOT + lane * 8);
  float* op = acc + (size_t)d * DTOT + lane * 8;
  float4 v0 = wp[0], v1 = wp[1];
  atomicAdd(op + 0, v0.x * aw); atomicAdd(op + 1, v0.y * aw);
  atomicAdd(op + 2, v0.z * aw); atomicAdd(op + 3, v0.w * aw);
  atomicAdd(op + 4, v1.x * aw); atomicAdd(op + 5, v1.y * aw);
  atomicAdd(op + 6, v1.z * aw); atomicAdd(op + 7, v1.w * aw);
}

// ---------------------------------------------------------------------------
// Single-destination dst-grouped softmax (p2s / a2s): ft[t] accumulated via
// pre-scaled partials. Phases 1/2 (max, denom) redundantly per block (cheap).
// ---------------------------------------------------------------------------
__global__ __launch_bounds__(256) void state_agg_kernel(
    const float* __restrict__ Wh, const float* __restrict__ snode,
    const float* __restrict__ dhead, const int* __restrict__ esrc,
    int Ne, float* __restrict__ ft) {
  __shared__ float red[256];
  __shared__ float mh[4], dh[4];
  int t = threadIdx.x;
  float d0 = dhead[0], d1 = dhead[1], d2 = dhead[2], d3 = dhead[3];

  float lm0 = -3.0e38f, lm1 = -3.0e38f, lm2 = -3.0e38f, lm3 = -3.0e38f;
  for (int i = t; i < Ne; i += 256) {
    int s = esrc[i];
    lm0 = fmaxf(lm0, leaky(snode[s * 4 + 0] + d0));
    lm1 = fmaxf(lm1, leaky(snode[s * 4 + 1] + d1));
    lm2 = fmaxf(lm2, leaky(snode[s * 4 + 2] + d2));
    lm3 = fmaxf(lm3, leaky(snode[s * 4 + 3] + d3));
  }
  float lmv[4] = {lm0, lm1, lm2, lm3};
  for (int h = 0; h < 4; ++h) {
    red[t] = lmv[h]; __syncthreads();
    for (int off = 128; off; off >>= 1) {
      if (t < off) red[t] = fmaxf(red[t], red[t + off]);
      __syncthreads();
    }
    if (t == 0) mh[h] = red[0];
    __syncthreads();
  }
  float m0 = mh[0], m1 = mh[1], m2 = mh[2], m3 = mh[3];

  float ls0 = 0.f, ls1 = 0.f, ls2 = 0.f, ls3 = 0.f;
  for (int i = t; i < Ne; i += 256) {
    int s = esrc[i];
    ls0 += __expf(leaky(snode[s * 4 + 0] + d0) - m0);
    ls1 += __expf(leaky(snode[s * 4 + 1] + d1) - m1);
    ls2 += __expf(leaky(snode[s * 4 + 2] + d2) - m2);
    ls3 += __expf(leaky(snode[s * 4 + 3] + d3) - m3);
  }
  float lsv[4] = {ls0, ls1, ls2, ls3};
  for (int h = 0; h < 4; ++h) {
    red[t] = lsv[h]; __syncthreads();
    for (int off = 128; off; off >>= 1) {
      if (t < off) red[t] += red[t + off];
      __syncthreads();
    }
    if (t == 0) dh[h] = red[0];
    __syncthreads();
  }

  // phase 3: slot t = (h, d); this block handles a strided slice of edges
  int h = t >> 6;
  float m = mh[h], invden = 1.f / dh[h];
  float dv = (h == 0) ? d0 : (h == 1) ? d1 : (h == 2) ? d2 : d3;
  float accv = 0.f;
  for (int i = blockIdx.x; i < Ne; i += gridDim.x) {
    int s = esrc[i];
    float a = __expf(leaky(snode[s * 4 + h] + dv) - m);
    accv += Wh[(size_t)s * DTOT + t] * a;
  }
  atomicAdd(ft + t, accv * invden);
}

// ---------------------------------------------------------------------------
// out = [relu(Whp+accP) | relu(Wha+accA) | relu(Whin+ftp+fta)]
// ---------------------------------------------------------------------------
__global__ __launch_bounds__(256) void finalize_kernel(
    const float* __restrict__ Whp, const float* __restrict__ accP,
    const float* __restrict__ Wha, const float* __restrict__ accA,
    const float* __restrict__ Whin, const float* __restrict__ ftp,
    const float* __restrict__ fta, float* __restrict__ out, int NP, int NA) {
  int i = blockIdx.x * 256 + threadIdx.x;
  int n1 = NP * DTOT, n2 = n1 + NA * DTOT;
  if (i < n1) {
    out[i] = fmaxf(Whp[i] + accP[i], 0.f);
  } else if (i < n2) {
    int j = i - n1;
    out[i] = fmaxf(Wha[j] + accA[j], 0.f);
  } else if (i < n2 + DTOT) {
    int j = i - n2;
    out[i] = fmaxf(Whin[j] + ftp[j] + fta[j], 0.f);
  }
}

// ---------------------------------------------------------------------------
extern "C" void kernel_launch(void* const* d_in, const int* in_sizes, int n_in,
                              void* d_out, int out_size, void* d_ws, size_t ws_size,
                              hipStream_t stream) {
  (void)n_in; (void)out_size; (void)ws_size;
  const float* feat_P = (const float*)d_in[0];
  const float* feat_A = (const float*)d_in[1];
  const float* feat_S = (const float*)d_in[2];
  const float* W_P   = (const float*)d_in[3];  const float* b_P   = (const float*)d_in[4];
  const float* W_A   = (const float*)d_in[5];  const float* b_A   = (const float*)d_in[6];
  const float* W_p2p = (const float*)d_in[7];  const float* b_p2p = (const float*)d_in[8];
  const float* W_p2a = (const float*)d_in[9];  const float* b_p2a = (const float*)d_in[10];
  const float* W_a2p = (const float*)d_in[11]; const float* b_a2p = (const float*)d_in[12];
  const float* W_a2a = (const float*)d_in[13]; const float* b_a2a = (const float*)d_in[14];
  const float* W_p2s = (const float*)d_in[15]; const float* b_p2s = (const float*)d_in[16];
  const float* W_a2s = (const float*)d_in[17]; const float* b_a2s = (const float*)d_in[18];
  const float* W_in  = (const float*)d_in[19]; const float* b_in  = (const float*)d_in[20];
  const float* att_p2p_s = (const float*)d_in[21]; const float* att_p2p_d = (const float*)d_in[22];
  const float* att_p2a_s = (const float*)d_in[23]; const float* att_p2a_d = (const float*)d_in[24];
  const float* att_a2p_s = (const float*)d_in[25]; const float* att_a2p_d = (const float*)d_in[26];
  const float* att_a2a_s = (const float*)d_in[27]; const float* att_a2a_d = (const float*)d_in[28];
  const float* att_p2s_s = (const float*)d_in[29]; const float* att_p2s_d = (const float*)d_in[30];
  const float* att_a2s_s = (const float*)d_in[31]; const float* att_a2s_d = (const float*)d_in[32];
  const int* e_p2p_s = (const int*)d_in[33]; const int* e_p2p_d = (const int*)d_in[34];
  const int* e_p2a_s = (const int*)d_in[35]; const int* e_p2a_d = (const int*)d_in[36];
  const int* e_a2p_s = (const int*)d_in[37]; const int* e_a2p_d = (const int*)d_in[38];
  const int* e_a2a_s = (const int*)d_in[39]; const int* e_a2a_d = (const int*)d_in[40];
  const int* e_p2s_s = (const int*)d_in[41];
  const int* e_a2s_s = (const int*)d_in[43];

  const int NP = in_sizes[0] / DIN;  // 4096
  const int NA = in_sizes[1] / DIN;  // 4096
  const int E  = in_sizes[33];       // 131072
  const int Eps = in_sizes[41];      // 4096
  const int Eas = in_sizes[43];      // 4096

  float* ws = (float*)d_ws;
  size_t o = 0;
  float* Whp   = ws + o; o += (size_t)NP * DTOT;
  float* Wha   = ws + o; o += (size_t)NA * DTOT;
  float* Whp2s = ws + o; o += (size_t)NP * DTOT;
  float* Wha2s = ws + o; o += (size_t)NA * DTOT;
  float* Whin  = ws + o; o += DTOT;
  float* Whpp  = ws + o; o += (size_t)NP * DTOT;   // p2p projection of feat_P
  float* Whpa  = ws + o; o += (size_t)NP * DTOT;   // p2a projection of feat_P
  float* Whap  = ws + o; o += (size_t)NA * DTOT;   // a2p projection of feat_A
  float* Whaa  = ws + o; o += (size_t)NA * DTOT;   // a2a projection of feat_A
  float* accP  = ws + o; o += (size_t)NP * DTOT;
  float* accA  = ws + o; o += (size_t)NA * DTOT;
  float* s_p2p = ws + o; o += (size_t)NP * HH;
  float* d_p2p = ws + o; o += (size_t)NP * HH;
  float* s_p2a = ws + o; o += (size_t)NP * HH;
  float* d_p2a = ws + o; o += (size_t)NA * HH;
  float* s_a2p = ws + o; o += (size_t)NA * HH;
  float* d_a2p = ws + o; o += (size_t)NP * HH;
  float* s_a2a = ws + o; o += (size_t)NA * HH;
  float* d_a2a = ws + o; o += (size_t)NA * HH;
  float* s_p2s = ws + o; o += (size_t)NP * HH;
  float* s_a2s = ws + o; o += (size_t)NA * HH;
  float* d_p2s = ws + o; o += 16;
  float* d_a2s = ws + o; o += 16;
  float* ftp2s = ws + o; o += DTOT;
  float* fta2s = ws + o; o += DTOT;

  hipMemsetAsync(accP, 0, (size_t)NP * DTOT * sizeof(float), stream);
  hipMemsetAsync(accA, 0, (size_t)NA * DTOT * sizeof(float), stream);
  hipMemsetAsync(ftp2s, 0, DTOT * sizeof(float), stream);
  hipMemsetAsync(fta2s, 0, DTOT * sizeof(float), stream);

  // ---- projections (WMMA fp32 GEMMs) ----
  auto gblk = [](int M) { return (((M + 15) / 16) * 16 + 7) / 8; };
  gemm_wmma_f32<<<gblk(NP), 256, 0, stream>>>(feat_P, W_P,   b_P,   Whp,   NP);
  gemm_wmma_f32<<<gblk(NA), 256, 0, stream>>>(feat_A, W_A,   b_A,   Wha,   NA);
  gemm_wmma_f32<<<gblk(NP), 256, 0, stream>>>(feat_P, W_p2s, b_p2s, Whp2s, NP);
  gemm_wmma_f32<<<gblk(NA), 256, 0, stream>>>(feat_A, W_a2s, b_a2s, Wha2s, NA);
  gemm_wmma_f32<<<gblk(1),  256, 0, stream>>>(feat_S, W_in,  b_in,  Whin,  1);
  gemm_wmma_f32<<<gblk(NP), 256, 0, stream>>>(feat_P, W_p2p, b_p2p, Whpp,  NP);
  gemm_wmma_f32<<<gblk(NP), 256, 0, stream>>>(feat_P, W_p2a, b_p2a, Whpa,  NP);
  gemm_wmma_f32<<<gblk(NA), 256, 0, stream>>>(feat_A, W_a2p, b_a2p, Whap,  NA);
  gemm_wmma_f32<<<gblk(NA), 256, 0, stream>>>(feat_A, W_a2a, b_a2a, Whaa,  NA);

  // ---- attention scalars per node ----
  auto sblk = [](int N) { return (N * HH + 255) / 256; };
  scores_kernel<<<sblk(NP), 256, 0, stream>>>(Whpp,  att_p2p_s, s_p2p, NP);
  scores_kernel<<<sblk(NP), 256, 0, stream>>>(Whp,   att_p2p_d, d_p2p, NP);
  scores_kernel<<<sblk(NP), 256, 0, stream>>>(Whpa,  att_p2a_s, s_p2a, NP);
  scores_kernel<<<sblk(NA), 256, 0, stream>>>(Wha,   att_p2a_d, d_p2a, NA);
  scores_kernel<<<sblk(NA), 256, 0, stream>>>(Whap,  att_a2p_s, s_a2p, NA);
  scores_kernel<<<sblk(NP), 256, 0, stream>>>(Whp,   att_a2p_d, d_a2p, NP);
  scores_kernel<<<sblk(NA), 256, 0, stream>>>(Whaa,  att_a2a_s, s_a2a, NA);
  scores_kernel<<<sblk(NA), 256, 0, stream>>>(Wha,   att_a2a_d, d_a2a, NA);
  scores_kernel<<<sblk(NP), 256, 0, stream>>>(Whp2s, att_p2s_s, s_p2s, NP);
  scores_kernel<<<sblk(1),  256, 0, stream>>>(Whin,  att_p2s_d, d_p2s, 1);
  scores_kernel<<<sblk(NA), 256, 0, stream>>>(Wha2s, att_a2s_s, s_a2s, NA);
  scores_kernel<<<sblk(1),  256, 0, stream>>>(Whin,  att_a2s_d, d_a2s, 1);

  // ---- edge aggregation (head-softmax + scatter-add) ----
  int eblk = (E + 7) / 8;
  edge_agg_kernel<<<eblk, 256, 0, stream>>>(Whpp, s_p2p, d_p2p, e_p2p_s, e_p2p_d, accP, E);
  edge_agg_kernel<<<eblk, 256, 0, stream>>>(Whpa, s_p2a, d_p2a, e_p2a_s, e_p2a_d, accA, E);
  edge_agg_kernel<<<eblk, 256, 0, stream>>>(Whap, s_a2p, d_a2p, e_a2p_s, e_a2p_d, accP, E);
  edge_agg_kernel<<<eblk, 256, 0, stream>>>(Whaa, s_a2a, d_a2a, e_a2a_s, e_a2a_d, accA, E);

  // ---- state aggregation (dst-grouped softmax, single destination) ----
  state_agg_kernel<<<8, 256, 0, stream>>>(Whp2s, s_p2s, d_p2s, e_p2s_s, Eps, ftp2s);
  state_agg_kernel<<<8, 256, 0, stream>>>(Wha2s, s_a2s, d_a2s, e_a2s_s, Eas, fta2s);

  // ---- finalize: relu + write outputs in tuple order ----
  int total = NP * DTOT + NA * DTOT + DTOT;
  finalize_kernel<<<(total + 255) / 256, 256, 0, stream>>>(
      Whp, accP, Wha, accA, Whin, ftp2s, fta2s, (float*)d_out, NP, NA);
}